// Detector_52836687675626
// MI455X (gfx1250) — compile-verified
//
#include <hip/hip_runtime.h>
#include <math.h>
#include <stdint.h>

#define NROI 2048
#define NCLS 21
#define NFG  20
#define TOPK 100
#define NT   256
#define CHUNK (NROI / NT)   // 8 elements per thread

// ---- CDNA5 async global->LDS path (guarded; fallback = plain LDS stores) ----
#if defined(__HIP_DEVICE_COMPILE__) && __has_builtin(__builtin_amdgcn_global_load_async_to_lds_b128)
#define HAVE_ASYNC 1
#else
#define HAVE_ASYNC 0
#endif

typedef int v4i __attribute__((ext_vector_type(4)));

#define GPTR(p) ((__attribute__((address_space(1))) v4i*)(uintptr_t)(p))
#define LPTR(p) ((__attribute__((address_space(3))) v4i*)(uint32_t)(uintptr_t)(p))

// ---------------------------------------------------------------------------
// Kernel 1: softmax-argmax + box decode.  decoded[n] = corner-format box.
// ---------------------------------------------------------------------------
__global__ __launch_bounds__(NT)
void decode_kernel(const float* __restrict__ rois,
                   const float* __restrict__ locs,
                   const float* __restrict__ scores,
                   float* __restrict__ decoded) {
    int n = blockIdx.x * NT + threadIdx.x;
    if (n >= NROI) return;

    const float* s = scores + n * NCLS;
    int best = 0;
    float bs = s[0];
#pragma unroll
    for (int c = 1; c < NCLS; ++c) {
        float v = s[c];
        if (v > bs) { bs = v; best = c; }   // strict > keeps first occurrence (argmax)
    }

    const float* L = locs + n * (NCLS * 4) + best * 4;
    float gx = L[0] / 10.0f, gy = L[1] / 10.0f;
    float gw = L[2] / 20.0f, gh = L[3] / 20.0f;

    float x0 = rois[n * 4 + 0], y0 = rois[n * 4 + 1];
    float x1 = rois[n * 4 + 2], y1 = rois[n * 4 + 3];
    float pcx = (x0 + x1) * 0.5f, pcy = (y0 + y1) * 0.5f;
    float pw = x1 - x0, ph = y1 - y0;

    float cx = gx * pw + pcx;
    float cy = gy * ph + pcy;
    float w  = expf(gw) * pw;
    float h  = expf(gh) * ph;

    decoded[n * 4 + 0] = cx - 0.5f * w;
    decoded[n * 4 + 1] = cy - 0.5f * h;
    decoded[n * 4 + 2] = cx + 0.5f * w;
    decoded[n * 4 + 3] = cy + 0.5f * h;
}

// ---------------------------------------------------------------------------
// Kernel 2: one block per foreground class.  Full on-chip pipeline:
//   async-stage boxes -> per-class softmax prob -> stable bitonic sort ->
//   gather sorted boxes+areas -> greedy NMS (forward-only, early-exit at
//   100 survivors) -> compact survivors into candidate table.
// ---------------------------------------------------------------------------
__global__ __launch_bounds__(NT)
void nms_kernel(const float* __restrict__ scores,
                const float* __restrict__ decoded,
                const float* __restrict__ min_score_p,
                const float* __restrict__ max_overlap_p,
                float* __restrict__ cand_score,
                float* __restrict__ cand_box,
                float* __restrict__ cand_label) {
    __shared__ float         skey[NROI];      // sort keys (class prob or -inf)
    __shared__ int           sidx[NROI];      // sorted -> original index; dead after
                                              // gather, overlaid below as sarea
    __shared__ float4        sbox[NROI];      // boxes in ORIGINAL order (async-staged)
    __shared__ float4        sboxs[NROI];     // boxes in SORTED order
    __shared__ unsigned char ssup[NROI];      // suppression flags (sorted order)
    __shared__ int           scnt[NT];
    __shared__ int           s_nv;

    float* sarea = (float*)sidx;              // overlay: areas in sorted order

    const int   tid = threadIdx.x;
    const int   cls = blockIdx.x + 1;         // foreground classes 1..20
    const float ms  = *min_score_p;
    const float thr = *max_overlap_p;
    const float NEG = -__builtin_inff();

    // Initialize this class's candidate slots (defaults match reference fill).
    if (tid < TOPK) {
        int o = blockIdx.x * TOPK + tid;
        cand_score[o]       = NEG;
        cand_box[o * 4 + 0] = 0.0f;
        cand_box[o * 4 + 1] = 0.0f;
        cand_box[o * 4 + 2] = 1.0f;
        cand_box[o * 4 + 3] = 1.0f;
        cand_label[o]       = 0.0f;
    }

    // Stage the 32KB decoded-box tile into LDS; overlaps softmax + sort below.
#if HAVE_ASYNC
    {
        const char* g = (const char*)decoded;
        char*       l = (char*)&sbox[0];
#pragma unroll
        for (int r = 0; r < CHUNK; ++r) {
            int i = r * NT + tid;
            __builtin_amdgcn_global_load_async_to_lds_b128(
                GPTR(g + (size_t)i * 16), LPTR(l + (size_t)i * 16), 0, 0);
        }
    }
#else
    for (int r = 0; r < CHUNK; ++r) {
        int i = r * NT + tid;
        sbox[i] = ((const float4*)decoded)[i];
    }
#endif

    // Per-ROI softmax prob of this class -> sort key (-inf if below threshold).
    for (int r = 0; r < CHUNK; ++r) {
        int i = r * NT + tid;
        const float* s = scores + i * NCLS;
        float m = s[0];
#pragma unroll
        for (int c = 1; c < NCLS; ++c) m = fmaxf(m, s[c]);
        float sum = 0.0f;
#pragma unroll
        for (int c = 0; c < NCLS; ++c) sum += expf(s[c] - m);
        float p = expf(s[cls] - m) / sum;
        skey[i] = (p > ms) ? p : NEG;
        sidx[i] = i;
    }
    __syncthreads();

    // Bitonic sort: key descending, index ascending on ties (== stable argsort(-s)).
    for (int k = 2; k <= NROI; k <<= 1) {
        for (int j = k >> 1; j > 0; j >>= 1) {
            for (int t = tid; t < NROI; t += NT) {
                int x = t ^ j;
                if (x > t) {
                    float k0 = skey[t], k1 = skey[x];
                    int   i0 = sidx[t], i1 = sidx[x];
                    bool after = (k0 < k1) || (k0 == k1 && i0 > i1); // t belongs after x
                    bool desc  = ((t & k) == 0);
                    if (desc ? after : !after) {
                        skey[t] = k1; skey[x] = k0;
                        sidx[t] = i1; sidx[x] = i0;
                    }
                }
            }
            __syncthreads();
        }
    }

    // Valid entries form a prefix after the sort; count them.
    if (tid == 0) s_nv = 0;
    __syncthreads();
    {
        int c = 0;
        for (int r = 0; r < CHUNK; ++r) {
            int i = r * NT + tid;
            if (skey[i] != NEG) c++;
        }
        atomicAdd(&s_nv, c);
    }

    // Async-staged boxes must be resident before the gather pass below.
#if HAVE_ASYNC
#if __has_builtin(__builtin_amdgcn_s_wait_asynccnt)
    __builtin_amdgcn_s_wait_asynccnt(0);
#else
    asm volatile("s_wait_asynccnt 0" ::: "memory");
#endif
#endif
    __syncthreads();

    const int nv = s_nv;

    // Gather boxes into sorted order, precompute areas (overlay kills sidx),
    // and initialize suppression flags (invalid tail starts suppressed).
    for (int r = 0; r < CHUNK; ++r) {
        int i = r * NT + tid;
        int oi = sidx[i];                 // read before overlaid write (data dep)
        float4 b = sbox[oi];
        sboxs[i] = b;
        sarea[i] = (b.z - b.x) * (b.w - b.y);
        ssup[i]  = (i < nv) ? 0 : 1;
    }
    __syncthreads();

    // Greedy NMS: forward-only suppression (j > i), early exit once this class
    // has 100 survivors (later survivors can never reach the global top-100,
    // and flags for rows < limit are final once row i is passed).
    int found = 0;
    int limit = nv;
    for (int i = 0; i < nv; ++i) {
        if (ssup[i] == 0) {               // uniform across block
            if (++found >= TOPK) { limit = i + 1; break; }
            float4 bi = sboxs[i];
            float  ai = sarea[i];
            for (int jj = i + 1 + tid; jj < nv; jj += NT) {
                if (ssup[jj]) continue;
                float4 bj = sboxs[jj];
                float lx = fmaxf(bi.x, bj.x), ly = fmaxf(bi.y, bj.y);
                float rx = fminf(bi.z, bj.z), ry = fminf(bi.w, bj.w);
                float iw = fmaxf(rx - lx, 0.0f), ih = fmaxf(ry - ly, 0.0f);
                float inter = iw * ih;
                float iou = inter / (ai + sarea[jj] - inter);
                if (iou > thr) ssup[jj] = 1;
            }
        }
        __syncthreads();
    }
    __syncthreads();

    // Stable compaction of the (<= 100) survivors in rows [0, limit).
    {
        int base = tid * CHUNK;
        int c = 0;
        for (int r = 0; r < CHUNK; ++r) {
            int i = base + r;
            if (i < limit && !ssup[i]) c++;
        }
        scnt[tid] = c;
        __syncthreads();
        if (tid == 0) {
            int run = 0;
            for (int t = 0; t < NT; ++t) { int v = scnt[t]; scnt[t] = run; run += v; }
        }
        __syncthreads();
        int rank = scnt[tid];
        for (int r = 0; r < CHUNK; ++r) {
            int i = base + r;
            if (i < limit && !ssup[i]) {
                if (rank < TOPK) {
                    int o = blockIdx.x * TOPK + rank;
                    cand_score[o] = skey[i];
                    float4 b = sboxs[i];
                    cand_box[o * 4 + 0] = b.x;
                    cand_box[o * 4 + 1] = b.y;
                    cand_box[o * 4 + 2] = b.z;
                    cand_box[o * 4 + 3] = b.w;
                    cand_label[o] = (float)cls;
                }
                rank++;
            }
        }
    }
}

// ---------------------------------------------------------------------------
// Kernel 3: global top-100 over the 2000 candidates (pad to 2048, bitonic).
// Candidate index order equals the reference's flattened order, so the
// idx-ascending tie-break matches lax.top_k exactly.
// ---------------------------------------------------------------------------
__global__ __launch_bounds__(NT)
void topk_kernel(const float* __restrict__ cand_score,
                 const float* __restrict__ cand_box,
                 const float* __restrict__ cand_label,
                 float* __restrict__ out) {
    __shared__ float key[NROI];
    __shared__ int   idx[NROI];
    const int   tid = threadIdx.x;
    const float NEG = -__builtin_inff();

    for (int t = tid; t < NROI; t += NT) {
        key[t] = (t < NFG * TOPK) ? cand_score[t] : NEG;
        idx[t] = t;
    }
    __syncthreads();

    for (int k = 2; k <= NROI; k <<= 1) {
        for (int j = k >> 1; j > 0; j >>= 1) {
            for (int t = tid; t < NROI; t += NT) {
                int x = t ^ j;
                if (x > t) {
                    float k0 = key[t], k1 = key[x];
                    int   i0 = idx[t], i1 = idx[x];
                    bool after = (k0 < k1) || (k0 == k1 && i0 > i1);
                    bool desc  = ((t & k) == 0);
                    if (desc ? after : !after) {
                        key[t] = k1; key[x] = k0;
                        idx[t] = i1; idx[x] = i0;
                    }
                }
            }
            __syncthreads();
        }
    }

    if (tid < TOPK) {
        float s = key[tid];
        float b0 = 0.0f, b1 = 0.0f, b2 = 1.0f, b3 = 1.0f, lab = 0.0f, sv = 0.0f;
        if (s > NEG) {                      // finite (NaN impossible here)
            int c = idx[tid];
            b0 = cand_box[c * 4 + 0];
            b1 = cand_box[c * 4 + 1];
            b2 = cand_box[c * 4 + 2];
            b3 = cand_box[c * 4 + 3];
            lab = cand_label[c];
            sv  = s;
        }
        out[tid * 4 + 0] = b0;
        out[tid * 4 + 1] = b1;
        out[tid * 4 + 2] = b2;
        out[tid * 4 + 3] = b3;
        out[4 * TOPK + tid]        = lab;  // labels as float
        out[4 * TOPK + TOPK + tid] = sv;   // scores
    }
}

// ---------------------------------------------------------------------------
extern "C" void kernel_launch(void* const* d_in, const int* in_sizes, int n_in,
                              void* d_out, int out_size, void* d_ws, size_t ws_size,
                              hipStream_t stream) {
    (void)in_sizes; (void)n_in; (void)out_size; (void)ws_size;

    const float* rois        = (const float*)d_in[0];
    const float* locs        = (const float*)d_in[1];
    const float* scores      = (const float*)d_in[2];
    const float* min_score   = (const float*)d_in[3];
    const float* max_overlap = (const float*)d_in[4];
    // d_in[5] = top_k (fixed at 100, baked into output layout)

    float* ws = (float*)d_ws;
    float* decoded    = ws;                      // 8192 floats
    float* cand_score = ws + 8192;               // 2000 floats
    float* cand_box   = ws + 8192 + 2048;        // 8000 floats
    float* cand_label = ws + 8192 + 2048 + 8192; // 2000 floats

    decode_kernel<<<NROI / NT, NT, 0, stream>>>(rois, locs, scores, decoded);
    nms_kernel<<<NFG, NT, 0, stream>>>(scores, decoded, min_score, max_overlap,
                                       cand_score, cand_box, cand_label);
    topk_kernel<<<1, NT, 0, stream>>>(cand_score, cand_box, cand_label, (float*)d_out);
}